// DMNlayer_6193342841089
// MI455X (gfx1250) — compile-verified
//
#include <hip/hip_runtime.h>
#include <hip/hip_bf16.h>
#include <stdint.h>

// Problem dims (fixed by the reference)
#define Q_DIM 2048
#define U_DIM 512
#define D_DIM 128
// Tile config
#define BQ 64
#define BU 64
#define PAD 4
#define LSTR (BU + PAD)      // 68 floats: keeps b128 LDS reads 16B-aligned, kills bank conflicts
#define NTHREADS 256

typedef float f32x4 __attribute__((ext_vector_type(4)));

#if defined(__has_builtin)
#if __has_builtin(__builtin_amdgcn_global_load_async_to_lds_b32) && \
    __has_builtin(__builtin_amdgcn_s_wait_asynccnt)
#define USE_ASYNC 1
#endif
#endif
#ifndef USE_ASYNC
#define USE_ASYNC 0
#endif

// Stage a 64-row x 128-col (row-major) global tile into LDS transposed as [128][LSTR].
// Lanes with consecutive tid cover consecutive d -> 128B coalesced global reads.
__device__ __forceinline__ void stage_tile(const float* __restrict__ g,
                                           float* lds, int tid) {
  const int d  = tid & 127;      // column (constant across k)
  const int r0 = tid >> 7;       // 0 or 1; row advances by 2 each k
#if USE_ASYNC
  __attribute__((address_space(1))) char* src =
      (__attribute__((address_space(1))) char*)(g + (size_t)r0 * D_DIM + d);
  __attribute__((address_space(3))) char* dst =
      (__attribute__((address_space(3))) char*)(lds + (size_t)d * LSTR + r0);
#pragma unroll
  for (int k = 0; k < 32; ++k) {
    __builtin_amdgcn_global_load_async_to_lds_b32(
        (__attribute__((address_space(1))) int*)src,
        (__attribute__((address_space(3))) int*)dst, 0, 0);
    src += 2 * D_DIM * sizeof(float);
    dst += 2 * sizeof(float);
  }
#else
#pragma unroll
  for (int k = 0; k < 32; ++k) {
    const int r = r0 + 2 * k;
    lds[(size_t)d * LSTR + r] = g[(size_t)r * D_DIM + d];
  }
#endif
}

__global__ __launch_bounds__(NTHREADS)
void dmn_minplus_kernel(const float* __restrict__ x,
                        const float* __restrict__ wmin,
                        const float* __restrict__ wmax,
                        float* __restrict__ out) {
  __shared__ float xs[D_DIM][LSTR];   // x tile, transposed [d][q]
  __shared__ float wn[D_DIM][LSTR];   // Wmin tile, transposed [d][u]
  __shared__ float wx[D_DIM][LSTR];   // Wmax tile, transposed [d][u]

  const int tid   = threadIdx.x;
  const int ubase = blockIdx.x * BU;
  const int qbase = blockIdx.y * BQ;

  stage_tile(x    + (size_t)qbase * D_DIM, &xs[0][0], tid);
  stage_tile(wmin + (size_t)ubase * D_DIM, &wn[0][0], tid);
  stage_tile(wmax + (size_t)ubase * D_DIM, &wx[0][0], tid);
#if USE_ASYNC
  __builtin_amdgcn_s_wait_asynccnt(0);
#endif
  __syncthreads();

  // tx -> u (so output b128 stores are coalesced across lanes), ty -> q
  const int tx = tid & 15;
  const int ty = tid >> 4;
  const int u0 = tx * 4;
  const int q0 = ty * 4;

  float acc[4][4];
#pragma unroll
  for (int i = 0; i < 4; ++i)
#pragma unroll
    for (int j = 0; j < 4; ++j) acc[i][j] = 3.402823466e+38f;

#pragma unroll 2
  for (int d = 0; d < D_DIM; ++d) {
    const f32x4 xv = *(const f32x4*)&xs[d][q0];  // broadcast within half-wave
    const f32x4 an = *(const f32x4*)&wn[d][u0];  // conflict-free across lanes
    const f32x4 ax = *(const f32x4*)&wx[d][u0];
#pragma unroll
    for (int i = 0; i < 4; ++i) {
      const float xq = xv[i];
#pragma unroll
      for (int j = 0; j < 4; ++j) {
        const float t1 = xq - an[j];      // x - Wmin
        const float t2 = ax[j] - xq;      // Wmax - x
        acc[i][j] = __builtin_fminf(__builtin_fminf(acc[i][j], t1), t2); // v_min3
      }
    }
  }

#pragma unroll
  for (int i = 0; i < 4; ++i) {
    f32x4 o = { acc[i][0], acc[i][1], acc[i][2], acc[i][3] };
    *(f32x4*)&out[(size_t)(qbase + q0 + i) * U_DIM + (ubase + u0)] = o;
  }
}

extern "C" void kernel_launch(void* const* d_in, const int* in_sizes, int n_in,
                              void* d_out, int out_size, void* d_ws, size_t ws_size,
                              hipStream_t stream) {
  (void)in_sizes; (void)n_in; (void)out_size; (void)d_ws; (void)ws_size;
  const float* x    = (const float*)d_in[0];
  const float* wmin = (const float*)d_in[1];
  const float* wmax = (const float*)d_in[2];
  float* out = (float*)d_out;

  dim3 grid(U_DIM / BU, Q_DIM / BQ);   // (8, 32)
  dim3 block(NTHREADS);
  dmn_minplus_kernel<<<grid, block, 0, stream>>>(x, wmin, wmax, out);
}